// _FallbackMamba_21157008900561
// MI455X (gfx1250) — compile-verified
//
#include <hip/hip_runtime.h>

#define D_MODEL 768
#define D_STATE 16
#define D_CONV  4
#define D_INNER 1536
#define BATCH   2
#define SEQ     4096
#define BT      (BATCH * SEQ)          // 8192 rows

typedef __attribute__((ext_vector_type(2))) float v2f;
typedef __attribute__((ext_vector_type(4))) float v4f;
typedef __attribute__((ext_vector_type(8))) float v8f;

__device__ __forceinline__ float silu_f(float v) {
    return v / (1.0f + __expf(-v));
}
__device__ __forceinline__ float softplus_f(float v) {
    return (v > 20.0f) ? v : log1pf(__expf(v));
}

// ---------------------------------------------------------------------------
// C[M,N] = act(A[M,K] @ W[N,K]^T + bias[N])   (all fp32, row-major)
//
// Block = 256 threads = 8 waves; block tile = 256 (M) x NT*16 (N).
// Wave w owns rows [m0+32w, m0+32w+32): 2 (M) x NT (N) WMMA 16x16 tiles.
// V_WMMA_F32_16X16X4_F32, K consumed in chunks of KC=32 (8 k-steps).
//
// W tile (shared by all 8 waves) is staged through double-buffered LDS with a
// 36-float row pitch (conflict-free b64 reads, 16B-aligned b128 stores).
// Steady-state chunk loop is branch-free (last chunk peeled): prefetch next W
// tile global->regs, compute current chunk (A global / B LDS, all NT B-frags
// loaded into distinct regs before the WMMA burst), regs->LDS, one barrier.
//
// NT: 4 when N % 64 == 0, 1 for the 16-wide B/C projections. act: 1=softplus.
// ---------------------------------------------------------------------------
template<int NT>
__global__ __launch_bounds__(256) void gemm_f32_wmma(
    const float* __restrict__ A, const float* __restrict__ W,
    const float* __restrict__ bias, float* __restrict__ C,
    int M, int N, int K, int act)
{
    constexpr int KC    = 32;            // k-chunk
    constexpr int PITCH = 36;            // LDS row pitch (floats)
    constexpr int NROWS = NT * 16;       // W rows per block
    constexpr int NQ    = NROWS * 8;     // float4s per W tile (KC=32 -> 8/row)
    constexpr int NITER = (NQ + 255) / 256;

    __shared__ float sW[2][NROWS * PITCH];

    const int tid  = threadIdx.x;
    const int wave = tid >> 5;
    const int lane = tid & 31;
    const int half = lane >> 4;          // K sub-pair {0,1} vs {2,3}
    const int r    = lane & 15;
    const int m0   = blockIdx.x * 256 + wave * 32;
    const int n0   = blockIdx.y * (NT * 16);

    v8f acc[2][NT];
#pragma unroll
    for (int mt = 0; mt < 2; ++mt)
#pragma unroll
        for (int nt = 0; nt < NT; ++nt)
            acc[mt][nt] = (v8f){0.f, 0.f, 0.f, 0.f, 0.f, 0.f, 0.f, 0.f};

    const float* Arow0 = A + (size_t)(m0 + r) * K + 2 * half;
    const float* Arow1 = Arow0 + (size_t)16 * K;

    auto compute_chunk = [&](int buf, int kc) {
#pragma unroll
        for (int kk = 0; kk < KC; kk += 4) {
            v2f a0 = *(const v2f*)(Arow0 + kc + kk);
            v2f a1 = *(const v2f*)(Arow1 + kc + kk);
            v2f bf[NT];
#pragma unroll
            for (int nt = 0; nt < NT; ++nt)
                bf[nt] = *(const v2f*)(
                    &sW[buf][(nt * 16 + r) * PITCH + 2 * half + kk]);
#pragma unroll
            for (int nt = 0; nt < NT; ++nt) {
                acc[0][nt] = __builtin_amdgcn_wmma_f32_16x16x4_f32(
                    false, a0, false, bf[nt], (short)0, acc[0][nt],
                    false, false);
                acc[1][nt] = __builtin_amdgcn_wmma_f32_16x16x4_f32(
                    false, a1, false, bf[nt], (short)0, acc[1][nt],
                    false, false);
            }
        }
    };

    // ---- prologue: stage chunk 0 into sW[0] -------------------------------
#pragma unroll
    for (int j = 0; j < NITER; ++j) {
        int i = tid + j * 256;
        if (NQ % 256 == 0 || i < NQ) {
            int row = i >> 3, q = i & 7;
            v4f g = *(const v4f*)(W + (size_t)(n0 + row) * K + q * 4);
            *(v4f*)(&sW[0][row * PITCH + q * 4]) = g;
        }
    }
    __syncthreads();

    const int nchunks = K / KC;
    int c = 0;
    for (; c < nchunks - 1; ++c) {       // steady state: branch-free body
        const int buf = c & 1;
        const int kc  = c * KC;

        // prefetch next W chunk into registers (overlaps the WMMA burst)
        v4f gr[NITER];
#pragma unroll
        for (int j = 0; j < NITER; ++j) {
            int i = tid + j * 256;
            if (NQ % 256 == 0 || i < NQ) {
                int row = i >> 3, q = i & 7;
                gr[j] = *(const v4f*)(W + (size_t)(n0 + row) * K +
                                      (kc + KC) + q * 4);
            }
        }

        compute_chunk(buf, kc);

        // commit prefetched chunk to the other LDS buffer
#pragma unroll
        for (int j = 0; j < NITER; ++j) {
            int i = tid + j * 256;
            if (NQ % 256 == 0 || i < NQ) {
                int row = i >> 3, q = i & 7;
                *(v4f*)(&sW[buf ^ 1][row * PITCH + q * 4]) = gr[j];
            }
        }
        __syncthreads();
    }
    compute_chunk(c & 1, c * KC);        // peeled last chunk

    // Epilogue. C/D layout: VGPR i -> row tile_m0 + i + 8*half, col tile_n0 + r
#pragma unroll
    for (int mt = 0; mt < 2; ++mt) {
#pragma unroll
        for (int nt = 0; nt < NT; ++nt) {
            const int col = n0 + nt * 16 + r;
            const float bv = bias ? bias[col] : 0.0f;
            const int rowbase = m0 + mt * 16 + half * 8;
#pragma unroll
            for (int i = 0; i < 8; ++i) {
                float v = acc[mt][nt][i] + bv;
                if (act == 1) v = softplus_f(v);
                C[(size_t)(rowbase + i) * N + col] = v;
            }
        }
    }
}

// ---------------------------------------------------------------------------
// Causal depthwise conv (taps=4, left pad 3) + bias + SiLU.
// Reads x_ssm_raw = xz[:, 0:D_INNER]; writes xs[BT, D_INNER].
// ---------------------------------------------------------------------------
__global__ __launch_bounds__(256) void conv_silu_kernel(
    const float* __restrict__ xz, const float* __restrict__ Wc,
    const float* __restrict__ bc, float* __restrict__ xs)
{
    size_t idx = (size_t)blockIdx.x * blockDim.x + threadIdx.x;
    if (idx >= (size_t)BT * D_INNER) return;
    const int d  = (int)(idx % D_INNER);
    const int bt = (int)(idx / D_INNER);
    const int t  = bt % SEQ;

    const float w0 = Wc[d * 4 + 0], w1 = Wc[d * 4 + 1];
    const float w2 = Wc[d * 4 + 2], w3 = Wc[d * 4 + 3];
    const float* base = xz + (size_t)bt * (2 * D_INNER) + d;
    const ptrdiff_t rs = 2 * D_INNER;

    float acc = bc[d] + w3 * base[0];
    if (t >= 1) acc += w2 * base[-1 * rs];
    if (t >= 2) acc += w1 * base[-2 * rs];
    if (t >= 3) acc += w0 * base[-3 * rs];
    xs[idx] = silu_f(acc);
}

// ---------------------------------------------------------------------------
// Selective scan. One thread per (batch, channel): 16 states in registers,
// A-row precomputed, B_t/C_t rows fetched as uniform-address b128 loads.
// Sequential over T (inherent); no barriers, no LDS.
// ---------------------------------------------------------------------------
__global__ __launch_bounds__(256) void scan_kernel(
    const float* __restrict__ xs, const float* __restrict__ dt,
    const float* __restrict__ Bt, const float* __restrict__ Ct,
    const float* __restrict__ A_log, float* __restrict__ y)
{
    const int d = blockIdx.x * 256 + threadIdx.x;   // channel
    const int b = blockIdx.y;                       // batch

    float Ad[D_STATE];
#pragma unroll
    for (int s = 0; s < D_STATE; ++s)
        Ad[s] = -__expf(A_log[(size_t)d * D_STATE + s]);

    float st[D_STATE];
#pragma unroll
    for (int s = 0; s < D_STATE; ++s) st[s] = 0.0f;

    size_t base   = (size_t)b * SEQ * D_INNER + d;
    size_t bcbase = (size_t)b * SEQ * D_STATE;

    for (int t = 0; t < SEQ; ++t) {
        const float dtt = dt[base];
        const float xt  = xs[base];
        v4f Bv[4], Cv[4];
#pragma unroll
        for (int q = 0; q < 4; ++q) {
            Bv[q] = *(const v4f*)(Bt + bcbase + 4 * q);
            Cv[q] = *(const v4f*)(Ct + bcbase + 4 * q);
        }
        const float dx = dtt * xt;
        float yt = 0.0f;
#pragma unroll
        for (int s = 0; s < D_STATE; ++s) {
            const float bs = Bv[s >> 2][s & 3];
            const float cs = Cv[s >> 2][s & 3];
            st[s] = st[s] * __expf(dtt * Ad[s]) + dx * bs;
            yt += st[s] * cs;
        }
        y[base] = yt;
        base   += D_INNER;
        bcbase += D_STATE;
    }
}

// ---------------------------------------------------------------------------
// y = (y + xs * D_skip[d]) * silu(z),  z = xz[:, D_INNER + d]. In place on y.
// ---------------------------------------------------------------------------
__global__ __launch_bounds__(256) void gate_kernel(
    const float* __restrict__ xz, const float* __restrict__ xs,
    const float* __restrict__ Dskip, float* __restrict__ y)
{
    size_t idx = (size_t)blockIdx.x * blockDim.x + threadIdx.x;
    if (idx >= (size_t)BT * D_INNER) return;
    const int d  = (int)(idx % D_INNER);
    const int bt = (int)(idx / D_INNER);
    const float z = xz[(size_t)bt * (2 * D_INNER) + D_INNER + d];
    y[idx] = (y[idx] + xs[idx] * Dskip[d]) * silu_f(z);
}

// ---------------------------------------------------------------------------
extern "C" void kernel_launch(void* const* d_in, const int* in_sizes, int n_in,
                              void* d_out, int out_size, void* d_ws, size_t ws_size,
                              hipStream_t stream)
{
    const float* x      = (const float*)d_in[0];
    const float* W_in   = (const float*)d_in[1];
    const float* W_conv = (const float*)d_in[2];
    const float* b_conv = (const float*)d_in[3];
    const float* W_dt   = (const float*)d_in[4];
    const float* b_dt   = (const float*)d_in[5];
    const float* A_log  = (const float*)d_in[6];
    const float* D_skip = (const float*)d_in[7];
    const float* W_B    = (const float*)d_in[8];
    const float* W_C    = (const float*)d_in[9];
    const float* W_out  = (const float*)d_in[10];
    float* out = (float*)d_out;

    float* ws = (float*)d_ws;
    size_t off = 0;
    float* xz  = ws + off; off += (size_t)BT * 2 * D_INNER;   // [8192, 3072]
    float* xs  = ws + off; off += (size_t)BT * D_INNER;       // [8192, 1536]
    float* dtb = ws + off; off += (size_t)BT * D_INNER;       // [8192, 1536]
    float* Btb = ws + off; off += (size_t)BT * D_STATE;       // [8192, 16]
    float* Ctb = ws + off; off += (size_t)BT * D_STATE;       // [8192, 16]
    float* yb  = ws + off; off += (size_t)BT * D_INNER;       // [8192, 1536]

    const size_t nElem = (size_t)BT * D_INNER;
    const int ewBlocks = (int)((nElem + 255) / 256);

    // 1. xz = x @ W_in^T                      [8192, 3072], K=768
    gemm_f32_wmma<4><<<dim3(BT / 256, (2 * D_INNER) / 64), 256, 0, stream>>>(
        x, W_in, nullptr, xz, BT, 2 * D_INNER, D_MODEL, 0);

    // 2. depthwise causal conv + SiLU -> xs
    conv_silu_kernel<<<ewBlocks, 256, 0, stream>>>(xz, W_conv, b_conv, xs);

    // 3. dt = softplus(xs @ W_dt^T + b_dt)    [8192, 1536], K=1536
    gemm_f32_wmma<4><<<dim3(BT / 256, D_INNER / 64), 256, 0, stream>>>(
        xs, W_dt, b_dt, dtb, BT, D_INNER, D_INNER, 1);

    // 4. B_t = xs @ W_B^T ; C_t = xs @ W_C^T  [8192, 16], K=1536
    gemm_f32_wmma<1><<<dim3(BT / 256, 1), 256, 0, stream>>>(
        xs, W_B, nullptr, Btb, BT, D_STATE, D_INNER, 0);
    gemm_f32_wmma<1><<<dim3(BT / 256, 1), 256, 0, stream>>>(
        xs, W_C, nullptr, Ctb, BT, D_STATE, D_INNER, 0);

    // 5. selective scan -> yb
    scan_kernel<<<dim3(D_INNER / 256, BATCH), 256, 0, stream>>>(
        xs, dtb, Btb, Ctb, A_log, yb);

    // 6. skip + gate (in place on yb)
    gate_kernel<<<ewBlocks, 256, 0, stream>>>(xz, xs, D_skip, yb);

    // 7. out = yb @ W_out^T                   [8192, 768], K=1536
    gemm_f32_wmma<4><<<dim3(BT / 256, D_MODEL / 64), 256, 0, stream>>>(
        yb, W_out, nullptr, out, BT, D_MODEL, D_INNER, 0);
}